// LipschitzNorm_37632503447976
// MI455X (gfx1250) — compile-verified
//
#include <hip/hip_runtime.h>
#include <stdint.h>

// LipschitzNorm for MI455X (gfx1250): pure HBM-streaming problem (~1.75 GB
// moved, ~0.8 GFLOP) -> ~75us roofline at 23.3 TB/s. No matmul structure, so
// the CDNA5-specific win is the data-movement path: b128 NT streaming loads,
// async global->LDS staging of the index slice (ASYNCcnt-tracked), and
// hardware global atomic max for the segment reduction.

typedef float v4f __attribute__((ext_vector_type(4)));

#define HEADS  8
#define NNODES 50000
#define TPB    256

__global__ void lips_init(float* __restrict__ node_max, int n) {
  int i = blockIdx.x * TPB + threadIdx.x;
  if (i < n) node_max[i] = 0.0f;  // norms are >= 0, so 0 is a valid -inf here
}

// L2 norm (sqrt of sum of squares) of 16 contiguous floats (one edge-head), non-temporal loads
__device__ __forceinline__ float norm16_nt(const float* __restrict__ base, int p) {
  const v4f* xp = (const v4f*)base + (size_t)p * 4;
  v4f a = __builtin_nontemporal_load(xp + 0);
  v4f b = __builtin_nontemporal_load(xp + 1);
  v4f c = __builtin_nontemporal_load(xp + 2);
  v4f d = __builtin_nontemporal_load(xp + 3);
  float s = 0.0f;
  s = fmaf(a.x, a.x, s); s = fmaf(a.y, a.y, s);
  s = fmaf(a.z, a.z, s); s = fmaf(a.w, a.w, s);
  s = fmaf(b.x, b.x, s); s = fmaf(b.y, b.y, s);
  s = fmaf(b.z, b.z, s); s = fmaf(b.w, b.w, s);
  s = fmaf(c.x, c.x, s); s = fmaf(c.y, c.y, s);
  s = fmaf(c.z, c.z, s); s = fmaf(c.w, c.w, s);
  s = fmaf(d.x, d.x, s); s = fmaf(d.y, d.y, s);
  s = fmaf(d.z, d.z, s); s = fmaf(d.w, d.w, s);
  return sqrtf(s);
}

// Stage this block's 32 destination-node ids into LDS with the CDNA5 async
// global->LDS copy engine (one b32 per lane of wave 0), then publish via
// s_wait_asynccnt + workgroup barrier. Avoids 8x-redundant index gathers.
__device__ __forceinline__ void stage_index_async(const int* __restrict__ index,
                                                  int e0, int E, int* s_idx) {
  int tid = threadIdx.x;
  if (tid < 32 && (e0 + tid) < E) {
    uint32_t lds_off = (uint32_t)(uintptr_t)(&s_idx[tid]);   // low 32b = LDS offset
    uint64_t gaddr   = (uint64_t)(uintptr_t)(index + e0 + tid);
    asm volatile("global_load_async_to_lds_b32 %0, %1, off"
                 :
                 : "v"(lds_off), "v"(gaddr)
                 : "memory");
  }
  asm volatile("s_wait_asynccnt 0" ::: "memory");  // only wave 0 has ASYNCcnt>0
  __syncthreads();
}

// Pass 1: nj = ||x_j[e,h,:]||, segment max into node_max[index[e]*H + h].
// nj >= 0 and node_max init = +0.0f, so IEEE bit pattern as signed int
// preserves ordering -> single hardware global_atomic_max per edge-head.
__global__ __launch_bounds__(TPB) void lips_pass1(const float* __restrict__ x_j,
                                                  const int* __restrict__ index,
                                                  float* __restrict__ node_max,
                                                  int E) {
  __shared__ int s_idx[TPB / HEADS];
  int tid = threadIdx.x;
  int p   = blockIdx.x * TPB + tid;          // edge-head pair id
  int e0  = blockIdx.x * (TPB / HEADS);      // first edge of this block
  stage_index_async(index, e0, E, s_idx);
  if (p < E * HEADS) {
    float nj = norm16_nt(x_j, p);
    int node = s_idx[tid >> 3];
    int h    = tid & (HEADS - 1);
    atomicMax((int*)node_max + node * HEADS + h, __float_as_int(nj));
  }
}

// Pass 2: ni = ||x_i||, gather node max (L2-resident 1.6MB), clamp, divide.
__global__ __launch_bounds__(TPB) void lips_pass2(const float* __restrict__ x_i,
                                                  const float* __restrict__ e_ij,
                                                  const int* __restrict__ index,
                                                  const float* __restrict__ node_max,
                                                  float* __restrict__ out,
                                                  int E) {
  __shared__ int s_idx[TPB / HEADS];
  int tid = threadIdx.x;
  int p   = blockIdx.x * TPB + tid;
  int e0  = blockIdx.x * (TPB / HEADS);
  stage_index_async(index, e0, E, s_idx);
  if (p < E * HEADS) {
    float ni = norm16_nt(x_i, p);
    int node = s_idx[tid >> 3];
    int h    = tid & (HEADS - 1);
    float mx = node_max[node * HEADS + h];        // hot in L2, keep RT policy
    float denom = fmaxf(4.0f * (ni + mx), 1e-12f);
    float e = __builtin_nontemporal_load(e_ij + p);
    e = fminf(fmaxf(e, -10000.0f), 10000.0f);
    __builtin_nontemporal_store(e / denom, out + p);
  }
}

extern "C" void kernel_launch(void* const* d_in, const int* in_sizes, int n_in,
                              void* d_out, int out_size, void* d_ws, size_t ws_size,
                              hipStream_t stream) {
  const float* e_ij  = (const float*)d_in[0];   // [E,H]
  const float* x_i   = (const float*)d_in[1];   // [E,H,16]
  const float* x_j   = (const float*)d_in[2];   // [E,H,16]
  const int*   index = (const int*)d_in[3];     // [E]

  int E  = in_sizes[3];
  int EH = E * HEADS;
  int NH = NNODES * HEADS;

  float* node_max = (float*)d_ws;               // 1.6 MB scratch, re-init per call

  lips_init<<<(NH + TPB - 1) / TPB, TPB, 0, stream>>>(node_max, NH);

  int blocks = (EH + TPB - 1) / TPB;            // 50,000 blocks for ref sizes
  lips_pass1<<<blocks, TPB, 0, stream>>>(x_j, index, node_max, E);
  lips_pass2<<<blocks, TPB, 0, stream>>>(x_i, e_ij, index, node_max,
                                         (float*)d_out, E);
}